// Model_65592740544618
// MI455X (gfx1250) — compile-verified
//
#include <hip/hip_runtime.h>
#include <hip/hip_bf16.h>

// MemN2N forward for MI455X (gfx1250, wave32).
// Hops are random-gather bound (VALU kernels); final u @ C[2]^T projection
// (M=32, K=64, N=100000) uses V_WMMA_F32_16X16X4_F32, one wave per 16-col
// tile producing all 32 rows so C[2] is streamed from HBM exactly once.

#define VOCAB 100000
#define EMBED 64
#define SENT  20
#define MEMS  50
#define BATCH 32

typedef __attribute__((ext_vector_type(2))) float v2f;
typedef __attribute__((ext_vector_type(8))) float v8f;

// ---------------------------------------------------------------------------
// u0[b,e] = sum_s A[queries[b,s], e] * enc[s, e]
// ---------------------------------------------------------------------------
__global__ __launch_bounds__(EMBED)
void query_embed_kernel(const int* __restrict__ queries,
                        const float* __restrict__ A,
                        const float* __restrict__ enc,
                        float* __restrict__ u) {
    const int b = blockIdx.x;
    const int e = threadIdx.x;
    float acc = 0.0f;
#pragma unroll
    for (int s = 0; s < SENT; ++s) {
        const int idx = queries[b * SENT + s];
        acc += A[(size_t)idx * EMBED + e] * enc[s * EMBED + e];
    }
    u[b * EMBED + e] = acc;
}

// ---------------------------------------------------------------------------
// Per (b, memory slot): gather+encode through both tables, and m . u dot.
// ---------------------------------------------------------------------------
__global__ __launch_bounds__(EMBED)
void hop_gather_dot_kernel(const int* __restrict__ stories,
                           const float* __restrict__ tabIn,
                           const float* __restrict__ tabC,
                           const float* __restrict__ enc,
                           const float* __restrict__ u,
                           float* __restrict__ m,
                           float* __restrict__ c,
                           float* __restrict__ dotted) {
    const int bm = blockIdx.x;          // 0 .. BATCH*MEMS-1
    const int b  = bm / MEMS;
    const int e  = threadIdx.x;         // 0..63
    const int* sent = stories + (size_t)bm * SENT;

    float ma = 0.0f, ca = 0.0f;
#pragma unroll
    for (int s = 0; s < SENT; ++s) {
        const int idx = sent[s];
        const float w = enc[s * EMBED + e];
        ma += tabIn[(size_t)idx * EMBED + e] * w;
        ca += tabC [(size_t)idx * EMBED + e] * w;
    }
    m[(size_t)bm * EMBED + e] = ma;
    c[(size_t)bm * EMBED + e] = ca;

    __shared__ float red[EMBED];
    red[e] = ma * u[b * EMBED + e];
    __syncthreads();
#pragma unroll
    for (int off = EMBED / 2; off > 0; off >>= 1) {
        if (e < off) red[e] += red[e + off];
        __syncthreads();
    }
    if (e == 0) dotted[bm] = red[0];
}

// ---------------------------------------------------------------------------
// Per batch: softmax over M=50, o = probs . c, u += o
// ---------------------------------------------------------------------------
__global__ __launch_bounds__(EMBED)
void softmax_update_kernel(const float* __restrict__ c,
                           const float* __restrict__ dotted,
                           float* __restrict__ u) {
    const int b = blockIdx.x;
    const int e = threadIdx.x;          // 0..63
    __shared__ float probs[MEMS];
    __shared__ float ssum;
    if (e == 0) {
        float mx = -3.4e38f;
        for (int i = 0; i < MEMS; ++i) mx = fmaxf(mx, dotted[b * MEMS + i]);
        float sm = 0.0f;
        for (int i = 0; i < MEMS; ++i) {
            const float ex = __expf(dotted[b * MEMS + i] - mx);
            probs[i] = ex;
            sm += ex;
        }
        ssum = sm;
    }
    __syncthreads();
    const float inv = 1.0f / ssum;
    float o = 0.0f;
#pragma unroll 5
    for (int i = 0; i < MEMS; ++i)
        o += probs[i] * c[((size_t)b * MEMS + i) * EMBED + e];
    u[b * EMBED + e] += o * inv;
}

// ---------------------------------------------------------------------------
// out[32, 100000] = u[32,64] @ C2[100000,64]^T via V_WMMA_F32_16X16X4_F32.
// One wave (32 lanes) per 16-column tile; two 16x16 accumulators cover all
// 32 batch rows while sharing the B-matrix (C2) loads.
//
// ISA operand layout (32-bit, wave32):
//   A 16x4 : lanes 0-15 -> M=lane,  VGPR0=K0, VGPR1=K1
//            lanes16-31 -> M=lane-16, VGPR0=K2, VGPR1=K3
//   B 4x16 : lanes 0-15 -> N=lane,  VGPR0=K0, VGPR1=K1
//            lanes16-31 -> N=lane-16, VGPR0=K2, VGPR1=K3
//   C/D 16x16: VGPR r -> M=r (lanes 0-15), M=r+8 (lanes 16-31); N=lane%16
// ---------------------------------------------------------------------------
__global__ __launch_bounds__(32)
void memn2n_out_gemm(const float* __restrict__ u,    // [32,64]
                     const float* __restrict__ C2,   // [100000,64]
                     float* __restrict__ out) {      // [32,100000]
    const int lane  = threadIdx.x;        // 0..31
    const int n     = lane & 15;          // column within tile / A row within tile
    const int khalf = (lane >> 4) << 1;   // 0 or 2
    const int v0    = blockIdx.x * 16;    // output column tile base

    const float* __restrict__ brow  = C2 + (size_t)(v0 + n) * EMBED;
    const float* __restrict__ arow0 = u + (size_t)n * EMBED;          // rows 0-15
    const float* __restrict__ arow1 = u + (size_t)(16 + n) * EMBED;   // rows 16-31

    v8f acc0 = {};
    v8f acc1 = {};

#pragma unroll
    for (int k0 = 0; k0 < EMBED; k0 += 4) {
        const int k = k0 + khalf;
        v2f a0, a1, bb;
        a0.x = arow0[k];     a0.y = arow0[k + 1];
        a1.x = arow1[k];     a1.y = arow1[k + 1];
        bb.x = brow[k];      bb.y = brow[k + 1];
        // (neg_a, A, neg_b, B, c_mod, C, reuse_a, reuse_b)
        acc0 = __builtin_amdgcn_wmma_f32_16x16x4_f32(false, a0, false, bb,
                                                     (short)0, acc0, false, false);
        acc1 = __builtin_amdgcn_wmma_f32_16x16x4_f32(false, a1, false, bb,
                                                     (short)0, acc1, false, false);
    }

    const int ncol  = v0 + n;
    const int rbase = (lane >> 4) << 3;   // 0 or 8
#pragma unroll
    for (int r = 0; r < 8; ++r) {
        out[(size_t)(rbase + r)      * VOCAB + ncol] = acc0[r];
        out[(size_t)(16 + rbase + r) * VOCAB + ncol] = acc1[r];
    }
}

// ---------------------------------------------------------------------------
extern "C" void kernel_launch(void* const* d_in, const int* in_sizes, int n_in,
                              void* d_out, int out_size, void* d_ws, size_t ws_size,
                              hipStream_t stream) {
    const int*   stories = (const int*)  d_in[0];   // [32,50,20]
    const int*   queries = (const int*)  d_in[1];   // [32,20]
    const float* A       = (const float*)d_in[2];   // [100000,64]
    const float* C       = (const float*)d_in[3];   // [3,100000,64]
    const float* enc     = (const float*)d_in[4];   // [20,64]
    float*       out     = (float*)d_out;           // [32,100000]

    float* ws     = (float*)d_ws;
    float* u      = ws;                                        // 2048 floats
    float* m      = ws + BATCH * EMBED;                        // 102400
    float* c      = m  + BATCH * MEMS * EMBED;                 // 102400
    float* dotted = c  + BATCH * MEMS * EMBED;                 // 1600

    query_embed_kernel<<<BATCH, EMBED, 0, stream>>>(queries, A, enc, u);

    for (int hop = 0; hop < 3; ++hop) {
        const float* tabIn = (hop == 0) ? A : C + (size_t)(hop - 1) * VOCAB * EMBED;
        const float* tabC  = C + (size_t)hop * VOCAB * EMBED;
        hop_gather_dot_kernel<<<BATCH * MEMS, EMBED, 0, stream>>>(
            stories, tabIn, tabC, enc, u, m, c, dotted);
        softmax_update_kernel<<<BATCH, EMBED, 0, stream>>>(c, dotted, u);
    }

    memn2n_out_gemm<<<VOCAB / 16, 32, 0, stream>>>(
        u, C + (size_t)2 * VOCAB * EMBED, out);
}